// ComplexDaubechiesWavelets_1288490189047
// MI455X (gfx1250) — compile-verified
//
#include <hip/hip_runtime.h>

typedef float v2f __attribute__((ext_vector_type(2)));
typedef float v8f __attribute__((ext_vector_type(8)));

#define SQ 0.70710678118654752440f
__device__ static const float cLR[6] = {-0.0662912607f*SQ, 0.1104854346f*SQ, 0.6629126074f*SQ,
                                         0.6629126074f*SQ, 0.1104854346f*SQ, -0.0662912607f*SQ};
__device__ static const float cLI[6] = {-0.0855816496f*SQ, -0.0855816496f*SQ, 0.1711632992f*SQ,
                                         0.1711632992f*SQ, -0.0855816496f*SQ, -0.0855816496f*SQ};
__device__ static const float cHR[6] = {-0.0662912607f*SQ, -0.1104854346f*SQ, 0.6629126074f*SQ,
                                        -0.6629126074f*SQ, 0.1104854346f*SQ, 0.0662912607f*SQ};
__device__ static const float cHI[6] = { 0.0855816496f*SQ, -0.0855816496f*SQ, -0.1711632992f*SQ,
                                         0.1711632992f*SQ, 0.0855816496f*SQ, -0.0855816496f*SQ};

// Packed filter-bank matrix F[16][24]:
//   M = 4*s + delta ; s: 0=lo_r 1=lo_i 2=hi_r 3=hi_i ; delta = output position 0..3
//   k < 12  : coefficient applied to real input sample m=k      (tap t = k - 2*delta)
//   k >= 12 : coefficient applied to imag input sample m=k-12   (complex combine folded in)
__device__ inline float acoef(int M, int k) {
  int s = M >> 2, d = M & 3;
  int kk = (k >= 12) ? (k - 12) : k;
  int t = kk - 2 * d;
  if (t < 0 || t > 5) return 0.0f;
  if (k < 12) return (s == 0) ? cLR[t] : (s == 1) ? cLI[t] : (s == 2) ? cHR[t] : cHI[t];
  return (s == 0) ? -cLI[t] : (s == 1) ? cLR[t] : (s == 2) ? -cHI[t] : cHR[t];
}

#define XP 48   // input tile pitch  (36 rows x 48 cols, cols>=36 zeroed)
#define YP 52   // intermediate pitch (bank-conflict free for transposed reads)
#define OPI 33  // output tile pitch

template<bool HAS_IMAG>
__global__ __launch_bounds__(128)
void dwt_level(const float* __restrict__ in_r, const float* __restrict__ in_i,
               long in_chstride, int in_pitch,
               float* __restrict__ ll_r, float* __restrict__ ll_i,
               long ll_chstride, int ll_pitch,
               float* __restrict__ out, int h, int w)
{
  __shared__ float Xr[36 * XP];
  __shared__ float Xi[HAS_IMAG ? 36 * XP : 1];
  __shared__ float Yr[32 * YP];
  __shared__ float Yi[32 * YP];
  __shared__ float O[2 * 32 * OPI];

  const int tid   = threadIdx.x;
  const int lane  = tid & 31;
  const int wv    = tid >> 5;
  const int laneM = lane & 15;
  const int half  = lane >> 4;

  const int ch = blockIdx.z;
  const int jb = blockIdx.y * 16;  // low-pass output row base
  const int cb = blockIdx.x * 16;  // low-pass output col base

  const float* inR = in_r + (size_t)ch * in_chstride;
  const float* inI = HAS_IMAG ? (in_i + (size_t)ch * in_chstride) : nullptr;

  // ---- stage 36x36 halo (rows 2jb-3..2jb+32, cols 2cb-3..2cb+32, circular) ----
  for (int idx = tid; idx < 36 * XP; idx += 128) {
    int m = idx / XP, n = idx % XP;
    float vr = 0.0f, vi = 0.0f;
    if (n < 36) {
      int gr = 2 * jb - 3 + m; gr += (gr < 0) ? h : 0; gr -= (gr >= h) ? h : 0;
      int gc = 2 * cb - 3 + n; gc += (gc < 0) ? w : 0; gc -= (gc >= w) ? w : 0;
      size_t off = (size_t)gr * in_pitch + gc;
      vr = inR[off];
      if (HAS_IMAG) vi = inI[off];
    }
    Xr[idx] = vr;
    if (HAS_IMAG) Xi[idx] = vi;
  }
  __syncthreads();

  // ---- per-lane constant A-matrix chunks (16x4 slices of F[16][24]) ----
  v2f A[6];
#pragma unroll
  for (int c = 0; c < 6; ++c) {
    int k = 4 * c + 2 * half;
    A[c].x = acoef(laneM, k);
    A[c].y = acoef(laneM, k + 1);
  }

  // ---- vertical pass: Y[32][36] (rows 0-15 lo, 16-31 hi), complex ----
  constexpr int NV = HAS_IMAG ? 6 : 3;  // level 0: xi == 0 -> K=12
  for (int u = wv; u < 12; u += 4) {
    int st = u % 3, g = u / 3;          // column strip, position group (j = 4g+delta)
    int col = 16 * st + laneM;
    v8f acc = {0.f, 0.f, 0.f, 0.f, 0.f, 0.f, 0.f, 0.f};
#pragma unroll
    for (int c = 0; c < NV; ++c) {
      v2f b;
      if (c < 3) {
        int m = 8 * g + 4 * c + 2 * half;
        b.x = Xr[m * XP + col];
        b.y = Xr[(m + 1) * XP + col];
      } else {
        int m = 8 * g + 4 * (c - 3) + 2 * half;
        b.x = Xi[m * XP + col];
        b.y = Xi[(m + 1) * XP + col];
      }
      acc = __builtin_amdgcn_wmma_f32_16x16x4_f32(false, A[c], false, b,
                                                  (short)0, acc, false, false);
    }
#pragma unroll
    for (int v = 0; v < 8; ++v) {        // D vgpr v: lanes0-15 M=v, lanes16-31 M=v+8
      int M = v + (half ? 8 : 0);
      int s = M >> 2, d = M & 3;
      int row = 4 * g + d + ((s & 2) ? 16 : 0);
      if (s & 1) Yi[row * YP + col] = acc[v];
      else       Yr[row * YP + col] = acc[v];
    }
  }
  __syncthreads();

  // ---- horizontal pass: O[2][32][32] (cols 0-15 lo, 16-31 hi) ----
  for (int u = wv; u < 8; u += 4) {
    int rs = u & 1, g = u >> 1;          // row strip (N), position group
    int n = 16 * rs + laneM;
    v8f acc = {0.f, 0.f, 0.f, 0.f, 0.f, 0.f, 0.f, 0.f};
#pragma unroll
    for (int c = 0; c < 6; ++c) {
      v2f b;
      if (c < 3) {
        int m = 8 * g + 4 * c + 2 * half;
        b.x = Yr[n * YP + m];
        b.y = Yr[n * YP + m + 1];
      } else {
        int m = 8 * g + 4 * (c - 3) + 2 * half;
        b.x = Yi[n * YP + m];
        b.y = Yi[n * YP + m + 1];
      }
      acc = __builtin_amdgcn_wmma_f32_16x16x4_f32(false, A[c], false, b,
                                                  (short)0, acc, false, false);
    }
#pragma unroll
    for (int v = 0; v < 8; ++v) {
      int M = v + (half ? 8 : 0);
      int s = M >> 2, d = M & 3;
      int ocol = ((s >> 1) ? 16 : 0) + 4 * g + d;
      int p = s & 1;
      O[p * 32 * OPI + n * OPI + ocol] = acc[v];
    }
  }
  __syncthreads();

  // ---- coalesced stores: quadrants -> d_out, LL -> next-level buffer ----
  const int OW = 1024;
  float* outR = out + (size_t)(2 * ch) * (OW * OW);
  float* outI = outR + (size_t)OW * OW;
  float* llR  = ll_r + (size_t)ch * ll_chstride;
  float* llI  = ll_i + (size_t)ch * ll_chstride;

  for (int q = tid; q < 512; q += 128) {
    int p    = q >> 8;
    int rem  = q & 255;
    int orow = rem >> 3;
    int oc4  = (rem & 7) * 4;
    const float* src = &O[p * 32 * OPI + orow * OPI + oc4];
    float4 val = make_float4(src[0], src[1], src[2], src[3]);
    if (orow < 16 && oc4 < 16) {
      float* dst = (p ? llI : llR) + (size_t)(jb + orow) * ll_pitch + (cb + oc4);
      *(float4*)dst = val;
    } else {
      int rg = (orow < 16) ? (jb + orow) : (h / 2 + jb + orow - 16);
      int cg = (oc4 < 16) ? (cb + oc4) : (w / 2 + cb + oc4 - 16);
      float* dst = (p ? outI : outR) + (size_t)rg * OW + cg;
      *(float4*)dst = val;
    }
  }
}

extern "C" void kernel_launch(void* const* d_in, const int* in_sizes, int n_in,
                              void* d_out, int out_size, void* d_ws, size_t ws_size,
                              hipStream_t stream) {
  (void)in_sizes; (void)n_in; (void)out_size; (void)ws_size;
  const float* images = (const float*)d_in[0];
  float* out = (float*)d_out;
  float* ws  = (float*)d_ws;
  const int NC = 24;  // 8 * 3 channels
  const size_t HW = (size_t)1024 * 1024;

  // ping-pong LL buffers in workspace (~63 MB total)
  float* aR = ws;
  float* aI = aR + (size_t)512 * 512 * NC;
  float* bR = aI + (size_t)512 * 512 * NC;
  float* bI = bR + (size_t)256 * 256 * NC;

  dim3 blk(128);
  // level 0 : imag input == 0
  dwt_level<false><<<dim3(32, 32, NC), blk, 0, stream>>>(
      images, nullptr, (long)HW, 1024, aR, aI, (long)512 * 512, 512, out, 1024, 1024);
  dwt_level<true><<<dim3(16, 16, NC), blk, 0, stream>>>(
      aR, aI, (long)512 * 512, 512, bR, bI, (long)256 * 256, 256, out, 512, 512);
  dwt_level<true><<<dim3(8, 8, NC), blk, 0, stream>>>(
      bR, bI, (long)256 * 256, 256, aR, aI, (long)128 * 128, 128, out, 256, 256);
  dwt_level<true><<<dim3(4, 4, NC), blk, 0, stream>>>(
      aR, aI, (long)128 * 128, 128, bR, bI, (long)64 * 64, 64, out, 128, 128);
  dwt_level<true><<<dim3(2, 2, NC), blk, 0, stream>>>(
      bR, bI, (long)64 * 64, 64, aR, aI, (long)32 * 32, 32, out, 64, 64);
  // level 5 : LL quadrant is final -> route LL pointers into d_out planes
  dwt_level<true><<<dim3(1, 1, NC), blk, 0, stream>>>(
      aR, aI, (long)32 * 32, 32, out, out + HW, (long)2 * HW, 1024, out, 32, 32);
}